// GAT_49735721287752
// MI455X (gfx1250) — compile-verified
//
#include <hip/hip_runtime.h>
#include <float.h>
#include <math.h>

#define N_NODES 100000
#define N_EDGES 1600000
#define N_TOT   (N_EDGES + N_NODES)
#define NEG_SLOPE 0.2f
#define SCAN_B 256

typedef __attribute__((ext_vector_type(2))) float v2f;
typedef __attribute__((ext_vector_type(8))) float v8f;

static inline int ceil_div(int a, int b) { return (a + b - 1) / b; }

__device__ __forceinline__ void edge_sd(const long long* __restrict__ ei,
                                        int e, int& s, int& d) {
    if (e < N_EDGES) { s = (int)ei[e]; d = (int)ei[N_EDGES + e]; }
    else             { s = e - N_EDGES; d = s; }
}

// ---------------------------------------------------------------------------
// H[n,F] = X[n,K] @ W[K,F]  (row-major), fp32 WMMA 16x16x4.
// One wave computes MT row-tiles x one col-tile (MT*16 x 16 output), loading
// each B fragment once per k-step and issuing MT WMMAs against it: B-matrix
// L2 traffic drops by MT. N_NODES = 80*1250, so MT=5 tiles exactly.
// ---------------------------------------------------------------------------
template<int K, int F, int MT>
__global__ void gemm_wmma_f32(const float* __restrict__ X,
                              const float* __restrict__ W,
                              float* __restrict__ H, int superRows) {
    const int wavesPerBlock = blockDim.x >> 5;
    int wave = blockIdx.x * wavesPerBlock + (threadIdx.x >> 5);
    const int colBlocks = F / 16;
    int rb = wave / colBlocks;
    int cb = wave % colBlocks;
    if (rb >= superRows) return;          // wave-uniform exit
    int lane = threadIdx.x & 31;
    int c    = lane & 15;
    int half = lane >> 4;

    const float* bcol = W + cb * 16 + c;
    const float* arow[MT];
    #pragma unroll
    for (int t = 0; t < MT; ++t)
        arow[t] = X + (size_t)((rb * MT + t) * 16 + c) * K + 2 * half;

    v8f acc[MT];
    #pragma unroll
    for (int t = 0; t < MT; ++t) acc[t] = {};

    #pragma unroll 4
    for (int k = 0; k < K; k += 4) {
        v2f b;
        b.x = bcol[(size_t)(k + 2 * half + 0) * F];
        b.y = bcol[(size_t)(k + 2 * half + 1) * F];
        #pragma unroll
        for (int t = 0; t < MT; ++t) {
            v2f a;
            a.x = arow[t][k + 0];
            a.y = arow[t][k + 1];
            acc[t] = __builtin_amdgcn_wmma_f32_16x16x4_f32(
                false, a, false, b, (short)0, acc[t], false, false);
        }
    }
    #pragma unroll
    for (int t = 0; t < MT; ++t) {
        float* orow = H + (size_t)((rb * MT + t) * 16 + 8 * half) * F + cb * 16 + c;
        #pragma unroll
        for (int j = 0; j < 8; ++j)
            orow[(size_t)j * F] = acc[t][j];
    }
}

// ---------------------------------------------------------------------------
// Wave-per-node: al_s[i] = h_i . a_s ; al_d[i] = h_i . a_d  (coalesced)
// ---------------------------------------------------------------------------
template<int F>
__global__ void node_alpha(const float* __restrict__ H,
                           const float* __restrict__ a_s,
                           const float* __restrict__ a_d,
                           float* __restrict__ als, float* __restrict__ ald) {
    int node = blockIdx.x * (blockDim.x >> 5) + (threadIdx.x >> 5);
    if (node >= N_NODES) return;
    int lane = threadIdx.x & 31;
    const float* h = H + (size_t)node * F;
    float s = 0.f, d = 0.f;
    #pragma unroll
    for (int f = lane; f < F; f += 32) {
        float v = h[f];
        s += v * a_s[f];
        d += v * a_d[f];
    }
    #pragma unroll
    for (int off = 16; off > 0; off >>= 1) {
        s += __shfl_xor(s, off, 32);
        d += __shfl_xor(d, off, 32);
    }
    if (lane == 0) { als[node] = s; ald[node] = d; }
}

// ---------------------------------------------------------------------------
// CSR build (once per call; shared by all 3 layers)
// ---------------------------------------------------------------------------
__global__ void zero_int(int* __restrict__ p, int n) {
    int t = blockIdx.x * blockDim.x + threadIdx.x;
    if (t < n) p[t] = 0;
}

__global__ void deg_count(const long long* __restrict__ ei, int* __restrict__ deg) {
    int e = blockIdx.x * blockDim.x + threadIdx.x;
    if (e >= N_TOT) return;
    int s, d; edge_sd(ei, e, s, d);
    atomicAdd(&deg[d], 1);
}

__global__ void scan_blocks(const int* __restrict__ deg, int* __restrict__ incl,
                            int* __restrict__ blockSums) {
    __shared__ int sh[SCAN_B];
    int t = threadIdx.x, g = blockIdx.x * SCAN_B + t;
    int v = (g < N_NODES) ? deg[g] : 0;
    sh[t] = v;
    __syncthreads();
    for (int off = 1; off < SCAN_B; off <<= 1) {
        int add = (t >= off) ? sh[t - off] : 0;
        __syncthreads();
        sh[t] += add;
        __syncthreads();
    }
    if (g < N_NODES) incl[g] = sh[t];
    if (t == SCAN_B - 1) blockSums[blockIdx.x] = sh[t];
}

__global__ void scan_sums(int* __restrict__ blockSums, int nb) {
    if (threadIdx.x == 0 && blockIdx.x == 0) {
        int run = 0;
        for (int i = 0; i < nb; ++i) { int v = blockSums[i]; blockSums[i] = run; run += v; }
    }
}

__global__ void scan_finalize(const int* __restrict__ deg, const int* __restrict__ incl,
                              const int* __restrict__ blockSums,
                              int* __restrict__ rowptr, int* __restrict__ cursor) {
    int g = blockIdx.x * blockDim.x + threadIdx.x;
    if (g < N_NODES) {
        int excl = incl[g] - deg[g] + blockSums[g / SCAN_B];
        rowptr[g] = excl;
        cursor[g] = excl;
    }
    if (g == 0) rowptr[N_NODES] = N_TOT;
}

__global__ void csr_fill(const long long* __restrict__ ei,
                         int* __restrict__ cursor, int* __restrict__ csr_src) {
    int e = blockIdx.x * blockDim.x + threadIdx.x;
    if (e >= N_TOT) return;
    int s, d; edge_sd(ei, e, s, d);
    int pos = atomicAdd(&cursor[d], 1);
    csr_src[pos] = s;
}

// ---------------------------------------------------------------------------
// Wave-per-node: softmax over incoming edges + weighted feature aggregation,
// fused bias (+ReLU). No float atomics; coalesced row gathers.
// ---------------------------------------------------------------------------
template<int F, int VPT, bool RELU>
__global__ void node_softmax_aggregate(const int* __restrict__ rowptr,
                                       const int* __restrict__ csr_src,
                                       const float* __restrict__ als,
                                       const float* __restrict__ ald,
                                       const float* __restrict__ H,
                                       float* __restrict__ elg,
                                       const float* __restrict__ bias,
                                       float* __restrict__ out) {
    int node = blockIdx.x * (blockDim.x >> 5) + (threadIdx.x >> 5);
    if (node >= N_NODES) return;        // wave-uniform
    int lane = threadIdx.x & 31;
    int start = rowptr[node], end = rowptr[node + 1];
    float aldi = ald[node];

    // pass 1: logits + max
    float mx = -FLT_MAX;
    for (int p = start + lane; p < end; p += 32) {
        int s = csr_src[p];
        float v = als[s] + aldi;
        v = (v > 0.f) ? v : NEG_SLOPE * v;
        elg[p] = v;
        mx = fmaxf(mx, v);
    }
    #pragma unroll
    for (int off = 16; off > 0; off >>= 1) mx = fmaxf(mx, __shfl_xor(mx, off, 32));

    // pass 2: exp + sum
    float sm = 0.f;
    for (int p = start + lane; p < end; p += 32) {
        float ex = __expf(elg[p] - mx);
        elg[p] = ex;
        sm += ex;
    }
    #pragma unroll
    for (int off = 16; off > 0; off >>= 1) sm += __shfl_xor(sm, off, 32);
    float inv = 1.0f / sm;              // self-loop guarantees sm > 0

    // pass 3: weighted aggregation, lane owns features [f0, f0+VPT);
    // unrolled x2 so two independent row gathers are in flight.
    float acc[VPT];
    #pragma unroll
    for (int j = 0; j < VPT; ++j) acc[j] = 0.f;
    int f0 = lane * VPT;
    bool active = (f0 < F);

    auto accum = [&](int p) {
        float a = elg[p] * inv;
        int s = csr_src[p];
        const float* hr = H + (size_t)s * F + f0;
        if (active) {
            if constexpr (VPT == 4) {
                float4 hv = *(const float4*)hr;
                acc[0] += a * hv.x; acc[1] += a * hv.y;
                acc[2] += a * hv.z; acc[3] += a * hv.w;
            } else if constexpr (VPT == 2) {
                float2 hv = *(const float2*)hr;
                acc[0] += a * hv.x; acc[1] += a * hv.y;
            } else {
                acc[0] += a * hr[0];
            }
        }
    };
    int p = start;
    for (; p + 1 < end; p += 2) { accum(p); accum(p + 1); }
    if (p < end) accum(p);

    if (active) {
        float* o = out + (size_t)node * F + f0;
        #pragma unroll
        for (int j = 0; j < VPT; ++j) {
            float v = acc[j] + bias[f0 + j];
            if (RELU) v = (v > 0.f) ? v : 0.f;
            o[j] = v;
        }
    }
}

// ---------------------------------------------------------------------------
extern "C" void kernel_launch(void* const* d_in, const int* in_sizes, int n_in,
                              void* d_out, int out_size, void* d_ws, size_t ws_size,
                              hipStream_t stream) {
    const float*     x    = (const float*)d_in[0];
    const long long* ei   = (const long long*)d_in[1];
    const float*     W1   = (const float*)d_in[2];
    const float*     as1  = (const float*)d_in[3];
    const float*     ad1  = (const float*)d_in[4];
    const float*     b1   = (const float*)d_in[5];
    const float*     W2   = (const float*)d_in[6];
    const float*     as2  = (const float*)d_in[7];
    const float*     ad2  = (const float*)d_in[8];
    const float*     b2   = (const float*)d_in[9];
    const float*     W3   = (const float*)d_in[10];
    const float*     as3  = (const float*)d_in[11];
    const float*     ad3  = (const float*)d_in[12];
    const float*     b3   = (const float*)d_in[13];
    float* out = (float*)d_out;

    // workspace carve-up
    float* h_buf   = (float*)d_ws;                        // N*128
    float* x_buf   = h_buf + (size_t)N_NODES * 128;       // N*128
    float* als     = x_buf + (size_t)N_NODES * 128;       // N
    float* ald     = als + N_NODES;                       // N
    float* elg     = ald + N_NODES;                       // N_TOT
    int*   deg     = (int*)(elg + N_TOT);                 // N
    int*   incl    = deg + N_NODES;                       // N
    int*   rowptr  = incl + N_NODES;                      // N+1
    int*   cursor  = rowptr + N_NODES + 1;                // N
    int*   blkSums = cursor + N_NODES;                    // nb
    int*   csr_src = blkSums + ceil_div(N_NODES, SCAN_B) + 1; // N_TOT

    const int BT = 256;
    constexpr int MT = 5;                                  // 80-row super-tiles
    const int superRows  = N_NODES / (16 * MT);            // 1250 exactly
    const int edgeBlocks = ceil_div(N_TOT, BT);
    const int nb = ceil_div(N_NODES, SCAN_B);
    const int nodeWaveBlocks = ceil_div(N_NODES, BT / 32); // 8 waves/block

    // ---- Build CSR (dst -> incoming src list), shared by all layers ----
    zero_int<<<ceil_div(N_NODES, BT), BT, 0, stream>>>(deg, N_NODES);
    deg_count<<<edgeBlocks, BT, 0, stream>>>(ei, deg);
    scan_blocks<<<nb, SCAN_B, 0, stream>>>(deg, incl, blkSums);
    scan_sums<<<1, 32, 0, stream>>>(blkSums, nb);
    scan_finalize<<<ceil_div(N_NODES, BT), BT, 0, stream>>>(deg, incl, blkSums, rowptr, cursor);
    csr_fill<<<edgeBlocks, BT, 0, stream>>>(ei, cursor, csr_src);

    // ---------------- Layer 1: 128 -> 128, ReLU ----------------
    {
        constexpr int K = 128, F = 128;
        int tiles = superRows * (F / 16);
        gemm_wmma_f32<K, F, MT><<<ceil_div(tiles, 4), 128, 0, stream>>>(x, W1, h_buf, superRows);
        node_alpha<F><<<nodeWaveBlocks, BT, 0, stream>>>(h_buf, as1, ad1, als, ald);
        node_softmax_aggregate<F, 4, true><<<nodeWaveBlocks, BT, 0, stream>>>(
            rowptr, csr_src, als, ald, h_buf, elg, b1, x_buf);
    }
    // ---------------- Layer 2: 128 -> 64, ReLU ----------------
    {
        constexpr int K = 128, F = 64;
        int tiles = superRows * (F / 16);
        gemm_wmma_f32<K, F, MT><<<ceil_div(tiles, 4), 128, 0, stream>>>(x_buf, W2, h_buf, superRows);
        node_alpha<F><<<nodeWaveBlocks, BT, 0, stream>>>(h_buf, as2, ad2, als, ald);
        node_softmax_aggregate<F, 2, true><<<nodeWaveBlocks, BT, 0, stream>>>(
            rowptr, csr_src, als, ald, h_buf, elg, b2, x_buf);
    }
    // ---------------- Layer 3: 64 -> 16, no ReLU, into d_out ----------------
    {
        constexpr int K = 64, F = 16;
        int tiles = superRows * (F / 16);
        gemm_wmma_f32<K, F, MT><<<ceil_div(tiles, 4), 128, 0, stream>>>(x_buf, W3, h_buf, superRows);
        node_alpha<F><<<nodeWaveBlocks, BT, 0, stream>>>(h_buf, as3, ad3, als, ald);
        node_softmax_aggregate<F, 1, false><<<nodeWaveBlocks, BT, 0, stream>>>(
            rowptr, csr_src, als, ald, h_buf, elg, b3, out);
    }
}